// BEVTextCLIPLoss_57397942944137
// MI455X (gfx1250) — compile-verified
//
#include <hip/hip_runtime.h>
#include <math.h>

typedef __attribute__((ext_vector_type(2))) float v2f;
typedef __attribute__((ext_vector_type(8))) float v8f;

#define B_SZ   16
#define C_GLB  512
#define C_LOC  256
#define N_TOK  4096   // 64*64

// ---- workspace layout (float offsets) ----
#define WS_INVB   0
#define WS_INVI   (WS_INVB  + B_SZ*N_TOK)
#define WS_INVP   (WS_INVI  + B_SZ*N_TOK)
#define WS_DIAGI  (WS_INVP  + B_SZ*N_TOK)
#define WS_DIAGP  (WS_DIAGI + B_SZ*N_TOK)
#define WS_BEVSUM (WS_DIAGP + B_SZ*N_TOK)
#define WS_POSI   (WS_BEVSUM + B_SZ*C_LOC)   // 256 per-block partials (16/batch)
#define WS_POSP   (WS_POSI + 256)
#define WS_NEG    (WS_POSP + 256)            // 256 per-block partials

__device__ __forceinline__ float sigm(float x) { return 1.0f / (1.0f + expf(-x)); }

// ---------------------------------------------------------------------------
// Kernel A: global + cross-modal contrast.
// One block, 4 waves. Phase 1: row inv-norms (48 rows). Phase 2: waves 0..2
// each compute one 16x16 logits tile via V_WMMA_F32_16X16X4_F32 over K=512
// (norms + exp(logit_scale) folded into the output tile). Phase 3: scalar
// softmax-CE on the three 16x16 tiles.
// ---------------------------------------------------------------------------
__global__ __launch_bounds__(128) void global_cross_kernel(
    const float* __restrict__ img, const float* __restrict__ pt,
    const float* __restrict__ txt, const float* __restrict__ lscale,
    float* __restrict__ out)
{
    __shared__ float inv[48];      // [img rows 0..15 | pt 16..31 | txt 32..47]
    __shared__ float lg[3 * 256];  // logits: [it | ip | pt@txt]

    const int tid = threadIdx.x;

    if (tid < 48) {
        const float* base = (tid < 16) ? img : (tid < 32 ? pt : txt);
        const float* p = base + (size_t)(tid & 15) * C_GLB;
        float ss = 0.0f;
        for (int k = 0; k < C_GLB; ++k) { float v = p[k]; ss += v * v; }
        inv[tid] = 1.0f / fmaxf(sqrtf(ss), 1e-12f);
    }
    __syncthreads();

    const int wave = tid >> 5;
    const int lane = tid & 31;

    if (wave < 3) {
        const float* X; const float* Y; int invx, invy;
        if (wave == 0)      { X = img; Y = txt; invx = 0;  invy = 32; }  // it
        else if (wave == 1) { X = img; Y = pt;  invx = 0;  invy = 16; }  // ip
        else                { X = pt;  Y = txt; invx = 16; invy = 32; }  // pt@txt

        // f32 WMMA operand layout: A 16x4 (lanes 0-15: K=0,1; 16-31: K=2,3),
        // B 4x16 mirrors it -> identical float2 fetch pattern for both.
        const int r  = lane & 15;
        const int kh = (lane >> 4) * 2;
        const float* xp = X + (size_t)r * C_GLB + kh;
        const float* yp = Y + (size_t)r * C_GLB + kh;

        v8f acc = {};
        #pragma unroll 4
        for (int i = 0; i < C_GLB / 4; ++i) {
            v2f a = *(const v2f*)(xp + 4 * i);
            v2f b = *(const v2f*)(yp + 4 * i);
            acc = __builtin_amdgcn_wmma_f32_16x16x4_f32(
                false, a, false, b, (short)0, acc, false, false);
        }

        // D layout: VGPR v, lane L -> M = v + 8*(L>>4), N = L&15.
        const float es = expf(lscale[0]);
        const int   n  = lane & 15;
        const int   mb = (lane >> 4) * 8;
        const float iy = inv[invy + n] * es;
        #pragma unroll
        for (int v = 0; v < 8; ++v) {
            int m = mb + v;
            lg[wave * 256 + m * 16 + n] = acc[v] * inv[invx + m] * iy;
        }
    }
    __syncthreads();

    if (tid == 0) {
        auto ceRow = [&](const float* L) {
            float tot = 0.0f;
            for (int r = 0; r < 16; ++r) {
                float mx = -1e30f;
                for (int c = 0; c < 16; ++c) mx = fmaxf(mx, L[r * 16 + c]);
                float s = 0.0f;
                for (int c = 0; c < 16; ++c) s += expf(L[r * 16 + c] - mx);
                tot += (logf(s) + mx) - L[r * 16 + r];
            }
            return tot / 16.0f;
        };
        auto ceCol = [&](const float* L) {
            float tot = 0.0f;
            for (int c = 0; c < 16; ++c) {
                float mx = -1e30f;
                for (int r = 0; r < 16; ++r) mx = fmaxf(mx, L[r * 16 + c]);
                float s = 0.0f;
                for (int r = 0; r < 16; ++r) s += expf(L[r * 16 + c] - mx);
                tot += (logf(s) + mx) - L[c * 16 + c];
            }
            return tot / 16.0f;
        };
        float ceRit = ceRow(lg),       ceCit = ceCol(lg);
        float ceRip = ceRow(lg + 256), ceCip = ceCol(lg + 256);
        float ceRpt = ceRow(lg + 512);
        out[0] = 0.25f * (ceRit + ceCit + ceRip + ceCip);  // GLOBAL_W = 1.0
        out[2] = 0.5f * (ceRit + ceRpt) * 0.5f;            // CROSS_W  = 0.5
    }
}

// ---------------------------------------------------------------------------
// Pass 1: per-token inv-norms + raw aligned dots. One token per lane
// (b32 loads, 128B contiguous per wave), 16 blocks/batch -> 2048 waves over
// the full tensor set for HBM latency hiding. Runs per 8-batch group so the
// group's img/pt/bev (~100 MB) stays resident in the 192 MB L2.
// ---------------------------------------------------------------------------
__global__ __launch_bounds__(256) void local_pass1(
    const float* __restrict__ imgf, const float* __restrict__ ptf,
    const float* __restrict__ bevf, float* __restrict__ ws, int b0)
{
    const int tid = threadIdx.x;
    const int b = b0 + blockIdx.y;
    const int n = blockIdx.x * 256 + tid;
    size_t off = (size_t)b * C_LOC * N_TOK + n;

    float ssi = 0.f, ssp = 0.f, ssb = 0.f, dib = 0.f, dpb = 0.f;
    #pragma unroll 8
    for (int c = 0; c < C_LOC; ++c) {
        float iv = imgf[off];
        float pv = ptf[off];
        float bv = bevf[off];
        ssi += iv * iv; ssp += pv * pv; ssb += bv * bv;
        dib += iv * bv; dpb += pv * bv;
        off += N_TOK;
    }
    float invi = 1.0f / fmaxf(sqrtf(ssi), 1e-12f);
    float invp = 1.0f / fmaxf(sqrtf(ssp), 1e-12f);
    float invb = 1.0f / fmaxf(sqrtf(ssb), 1e-12f);
    float di = dib * invi * invb;
    float dp = dpb * invp * invb;

    const int idx = b * N_TOK + n;
    ws[WS_INVB  + idx] = invb;
    ws[WS_INVI  + idx] = invi;
    ws[WS_INVP  + idx] = invp;
    ws[WS_DIAGI + idx] = di;
    ws[WS_DIAGP + idx] = dp;

    __shared__ float r1[256], r2[256];
    r1[tid] = di;
    r2[tid] = dp;
    __syncthreads();
    for (int s = 128; s > 0; s >>= 1) {
        if (tid < s) { r1[tid] += r1[tid + s]; r2[tid] += r2[tid + s]; }
        __syncthreads();
    }
    if (tid == 0) {
        int bf = b * 16 + blockIdx.x;
        ws[WS_POSI + bf] = r1[0];
        ws[WS_POSP + bf] = r2[0];
    }
}

// ---------------------------------------------------------------------------
// Pass 2: bev_sum[b,c] = sum_n bev_raw[b,c,n]*invb[b,n]. One block per (c,b);
// bev re-read hits L2 (same group as pass 1); invb (16 KB/batch) L2-resident.
// ---------------------------------------------------------------------------
__global__ __launch_bounds__(256) void local_pass2(
    const float* __restrict__ bevf, float* __restrict__ ws, int b0)
{
    const int tid = threadIdx.x;
    const int c = blockIdx.x, b = b0 + blockIdx.y;
    size_t off = ((size_t)b * C_LOC + c) * N_TOK + tid * 4;
    float4 bv = *(const float4*)(bevf + off);
    float4 ib = *(const float4*)(ws + WS_INVB + b * N_TOK + tid * 4);
    float acc = bv.x * ib.x + bv.y * ib.y + bv.z * ib.z + bv.w * ib.w;
    __shared__ float red[256];
    red[tid] = acc;
    __syncthreads();
    for (int s = 128; s > 0; s >>= 1) {
        if (tid < s) red[tid] += red[tid + s];
        __syncthreads();
    }
    if (tid == 0) ws[WS_BEVSUM + b * C_LOC + c] = red[0];
}

// ---------------------------------------------------------------------------
// Pass 3: row = inv * (raw_col . bev_sum); sigmoid(neg) partial sums.
// Same 1-token-per-lane layout as pass 1; img/pt re-reads hit L2.
// ---------------------------------------------------------------------------
__global__ __launch_bounds__(256) void local_pass3(
    const float* __restrict__ imgf, const float* __restrict__ ptf,
    float* __restrict__ ws, int b0)
{
    const int tid = threadIdx.x;
    const int b = b0 + blockIdx.y;
    __shared__ float bs[C_LOC];
    bs[tid] = ws[WS_BEVSUM + b * C_LOC + tid];
    __syncthreads();

    const int n = blockIdx.x * 256 + tid;
    size_t off = (size_t)b * C_LOC * N_TOK + n;
    float ai = 0.f, ap = 0.f;
    #pragma unroll 8
    for (int c = 0; c < C_LOC; ++c) {
        float s = bs[c];
        ai += imgf[off] * s;
        ap += ptf[off] * s;
        off += N_TOK;
    }
    const int idx = b * N_TOK + n;
    const float rn = 1.0f / (float)(N_TOK - 1);
    float ssum = sigm((ai * ws[WS_INVI + idx] - ws[WS_DIAGI + idx]) * rn)
               + sigm((ap * ws[WS_INVP + idx] - ws[WS_DIAGP + idx]) * rn);

    __shared__ float red[256];
    red[tid] = ssum;
    __syncthreads();
    for (int s = 128; s > 0; s >>= 1) {
        if (tid < s) red[tid] += red[tid + s];
        __syncthreads();
    }
    if (tid == 0) ws[WS_NEG + b * 16 + blockIdx.x] = red[0];
}

// ---------------------------------------------------------------------------
// Finalize: deterministic serial combine; writes out[1], out[3].
// ---------------------------------------------------------------------------
__global__ void finalize_kernel(const float* __restrict__ ws, float* __restrict__ out)
{
    if (threadIdx.x == 0 && blockIdx.x == 0) {
        float pos_loss = 0.0f;
        for (int b = 0; b < B_SZ; ++b) {
            float si = 0.0f, sp = 0.0f;
            for (int t = 0; t < 16; ++t) {
                si += ws[WS_POSI + b * 16 + t];
                sp += ws[WS_POSP + b * 16 + t];
            }
            pos_loss += -logf(sigm(si / (float)N_TOK) + 1e-8f);
            pos_loss += -logf(sigm(sp / (float)N_TOK) + 1e-8f);
        }
        pos_loss /= (float)(2 * B_SZ);
        float ns = 0.0f;
        for (int i = 0; i < 256; ++i) ns += ws[WS_NEG + i];
        float neg_loss = ns / (float)(2 * B_SZ * N_TOK);
        float l_local = 0.5f * (pos_loss + 0.1f * neg_loss);  // LOCAL_W = 0.5
        out[1] = l_local;
        out[3] = out[0] + out[1] + out[2];
    }
}

extern "C" void kernel_launch(void* const* d_in, const int* in_sizes, int n_in,
                              void* d_out, int out_size, void* d_ws, size_t ws_size,
                              hipStream_t stream) {
    const float* img_g = (const float*)d_in[0];
    const float* pt_g  = (const float*)d_in[1];
    const float* txt_g = (const float*)d_in[2];
    const float* img_f = (const float*)d_in[3];
    const float* pt_f  = (const float*)d_in[4];
    const float* bev_f = (const float*)d_in[5];
    const float* lsc   = (const float*)d_in[6];
    float* out = (float*)d_out;
    float* ws  = (float*)d_ws;

    global_cross_kernel<<<1, 128, 0, stream>>>(img_g, pt_g, txt_g, lsc, out);

    // Process batches in 2 groups of 8 so each group's ~100 MB working set
    // fits the 192 MB L2: pass2/pass3 re-reads become L2 hits, and total HBM
    // traffic drops from ~400 MB to ~201 MB (~9 us at 23.3 TB/s).
    for (int g = 0; g < 2; ++g) {
        int b0 = g * 8;
        local_pass1<<<dim3(16, 8), 256, 0, stream>>>(img_f, pt_f, bev_f, ws, b0);
        local_pass2<<<dim3(C_LOC, 8), 256, 0, stream>>>(bev_f, ws, b0);
        local_pass3<<<dim3(16, 8), 256, 0, stream>>>(img_f, pt_f, ws, b0);
    }
    finalize_kernel<<<1, 32, 0, stream>>>(ws, out);
}